// LLObjectCondensation_65627100283528
// MI455X (gfx1250) — compile-verified
//
#include <hip/hip_runtime.h>

#define BB 4
#define NN 100000
#define KK 128
#define TB ((NN + 127) / 128)   // 16-hit tiles per 8-wave block -> 128 hits/block

typedef float v2f __attribute__((ext_vector_type(2)));
typedef float v8f __attribute__((ext_vector_type(8)));

__device__ __forceinline__ float clipb(float b) {
    return fminf(fmaxf(b, 1e-4f), 1.0f - 1e-4f);
}

// ---------------- kernel 0: zero argmax table ----------------
__global__ void k_init(unsigned long long* packed) {
    int t = blockIdx.x * 256 + threadIdx.x;
    if (t < BB * KK) packed[t] = 0ull;
}

// ---------------- kernel 1: per-hit pass ----------------
// segmented argmax via atomicMax of (qbits<<32 | ~n); per-block partial sums of
// [beta*noise, noise, p, p*huber, p*pos2, p*time2]
__global__ void k_hits(const float* __restrict__ beta, const float* __restrict__ pe,
                       const float* __restrict__ pp,   const float* __restrict__ pt,
                       const float* __restrict__ te,   const float* __restrict__ tp,
                       const float* __restrict__ tt,   const int* __restrict__ tidx,
                       unsigned long long* packed, float* __restrict__ pA) {
    int gid = blockIdx.x * 256 + threadIdx.x;
    float s0 = 0.f, s1 = 0.f, s2 = 0.f, s3 = 0.f, s4 = 0.f, s5 = 0.f;
    if (gid < BB * NN) {
        int b = gid / NN;
        int n = gid - b * NN;
        float be = clipb(beta[gid]);
        float ath = 0.5f * logf((1.0f + be) / (1.0f - be));
        float ath2 = ath * ath;
        float q = ath2 + 0.5f;                 // Q_MIN = 0.5
        int ti = tidx[gid];
        if (ti > 0) {
            unsigned long long key =
                ((unsigned long long)__float_as_uint(q) << 32) |
                (unsigned long long)(0xFFFFFFFFu - (unsigned)n);   // first-max tiebreak
            atomicMax(&packed[b * KK + (ti - 1)], key);
        }
        float noise = (ti == 0) ? 1.0f : 0.0f;
        float p = ath2 * (1.0f - noise);
        float e = pe[gid] - te[gid];
        float ae = fabsf(e);
        float hub = (ae <= 2.0f) ? 0.5f * e * e : 2.0f * (ae - 1.0f);  // HUBER_D=2
        float dx = pp[2 * gid] - tp[2 * gid];
        float dy = pp[2 * gid + 1] - tp[2 * gid + 1];
        float dt = pt[gid] - tt[gid];
        s0 = be * noise; s1 = noise; s2 = p;
        s3 = p * hub; s4 = p * (dx * dx + dy * dy); s5 = p * dt * dt;
    }
    __shared__ float red[256];
    float s[6] = {s0, s1, s2, s3, s4, s5};
    for (int j = 0; j < 6; ++j) {
        red[threadIdx.x] = s[j];
        __syncthreads();
        for (int st = 128; st > 0; st >>= 1) {
            if (threadIdx.x < st) red[threadIdx.x] += red[threadIdx.x + st];
            __syncthreads();
        }
        if (threadIdx.x == 0) pA[blockIdx.x * 6 + j] = red[0];
        __syncthreads();
    }
}

// ---------------- kernel 2: extract condensation points ----------------
__global__ void k_alpha(const float* __restrict__ beta, const float* __restrict__ cc,
                        const unsigned long long* __restrict__ packed,
                        float* __restrict__ objX, float* __restrict__ objY,
                        float* __restrict__ objQ, float* __restrict__ scalB) {
    int t = threadIdx.x;                 // 0..511 (B*K)
    int b = t / KK;
    unsigned long long pk = packed[t];
    float ex = 0.f, bA = 0.f, ox = 0.f, oy = 0.f, qa = 0.f;
    if (pk != 0ull) {
        ex = 1.0f;
        unsigned n = 0xFFFFFFFFu - (unsigned)(pk & 0xFFFFFFFFull);
        qa = __uint_as_float((unsigned)(pk >> 32));   // exact q of argmax hit
        int idx = b * NN + (int)n;
        bA = clipb(beta[idx]);
        ox = cc[2 * idx]; oy = cc[2 * idx + 1];
    }
    objX[t] = ox; objY[t] = oy; objQ[t] = qa * ex;    // q_alpha * exists
    __shared__ float r0[512], r1[512];
    r0[t] = ex * (1.0f - bA);
    r1[t] = ex;
    __syncthreads();
    for (int st = 256; st > 0; st >>= 1) {
        if (t < st) { r0[t] += r0[t + st]; r1[t] += r1[t + st]; }
        __syncthreads();
    }
    if (t == 0) { scalB[0] = r0[0]; scalB[1] = r1[0]; }
}

// ---------------- kernel 3: pairwise core via f32 WMMA ----------------
// dist^2[m,n] = A x B with A row = (hx, hy, |h|^2, 1), B col = (-2ox, -2oy, 1, |o|^2)
__global__ void k_pair(const float* __restrict__ beta, const float* __restrict__ cc,
                       const int* __restrict__ tidx,
                       const float* __restrict__ objX, const float* __restrict__ objY,
                       const float* __restrict__ objQ, float* __restrict__ pC) {
    int b    = blockIdx.x / TB;
    int tb   = blockIdx.x % TB;
    int tid  = threadIdx.x;
    int wave = tid >> 5;
    int lane = tid & 31;
    int m    = lane & 15;
    int hi   = lane >> 4;          // which K-half of the fragment this lane holds

    __shared__ float sX[KK], sY[KK], sQ[KK];
    if (tid < KK) {
        sX[tid] = objX[b * KK + tid];
        sY[tid] = objY[b * KK + tid];
        sQ[tid] = objQ[b * KK + tid];
    }
    __syncthreads();

    int hitBase = tb * 128 + wave * 16;
    int hit = hitBase + m;
    bool valid = hit < NN;
    float hx = 0.f, hy = 0.f, qh = 0.f;
    int oh = -2;
    if (valid) {
        int g = b * NN + hit;
        hx = cc[2 * g]; hy = cc[2 * g + 1];
        float be = clipb(beta[g]);
        float ath = 0.5f * logf((1.0f + be) / (1.0f - be));
        qh = ath * ath + 0.5f;
        oh = tidx[g] - 1;          // -1 = noise
    }
    float hh = hx * hx + hy * hy;
    v2f a;
    a.x = hi ? hh   : hx;          // lanes 0-15: K0,K1 ; lanes 16-31: K2,K3
    a.y = hi ? 1.0f : hy;

    // Row broadcasts are t-invariant: hoist all 16 bpermutes out of the tile loop.
    float qrow[8];
    int   orow[8];
#pragma unroll
    for (int r = 0; r < 8; ++r) {
        int src = r + 8 * hi;              // row M = r (+8 for upper lane-half)
        qrow[r] = __shfl(qh, src, 32);
        orow[r] = __shfl(oh, src, 32);
    }

    // Prebuild per-tile B fragments + object charge (pure registers in the loop).
    v2f  bfs[KK / 16];
    float qas[KK / 16];
#pragma unroll
    for (int t = 0; t < KK / 16; ++t) {
        int k = t * 16 + m;
        float ox = sX[k], oy = sY[k];
        float oo = ox * ox + oy * oy;
        bfs[t].x = hi ? 1.0f : (-2.0f * ox);
        bfs[t].y = hi ? oo   : (-2.0f * oy);
        qas[t] = sQ[k];
    }

    float att = 0.f, rep = 0.f;
#pragma unroll
    for (int t = 0; t < KK / 16; ++t) {
        int k = t * 16 + m;
        float qa = qas[t];
        v8f c = {0.f, 0.f, 0.f, 0.f, 0.f, 0.f, 0.f, 0.f};
        v8f d = __builtin_amdgcn_wmma_f32_16x16x4_f32(
            false, a, false, bfs[t], (short)0, c, false, false);
#pragma unroll
        for (int r = 0; r < 8; ++r) {
            float d2 = fmaxf(d[r], 0.0f) + 1e-12f;
            float dist = __builtin_amdgcn_sqrtf(d2);      // raw v_sqrt_f32, ~1ulp
            float qq = qrow[r] * qa;                      // includes exists + OOR mask
            float attc = qq * d2;
            float repc = qq * fmaxf(1.0f - dist, 0.0f);
            bool mem = (orow[r] == k);
            att += mem ? attc : 0.0f;                     // branchless: v_cndmask
            rep += mem ? 0.0f : repc;
        }
    }

    __shared__ float rA[256], rR[256];
    rA[tid] = att; rR[tid] = rep;
    __syncthreads();
    for (int st = 128; st > 0; st >>= 1) {
        if (tid < st) { rA[tid] += rA[tid + st]; rR[tid] += rR[tid + st]; }
        __syncthreads();
    }
    if (tid == 0) { pC[blockIdx.x * 2] = rA[0]; pC[blockIdx.x * 2 + 1] = rR[0]; }
}

// ---------------- kernel 4: final deterministic combine ----------------
__global__ void k_final(const float* __restrict__ pA, int nA,
                        const float* __restrict__ pC, int nC,
                        const float* __restrict__ scalB, float* __restrict__ out) {
    int tid = threadIdx.x;
    float s[8] = {0, 0, 0, 0, 0, 0, 0, 0};
    for (int i = tid; i < nA; i += 256)
        for (int j = 0; j < 6; ++j) s[j] += pA[i * 6 + j];
    for (int i = tid; i < nC; i += 256) {
        s[6] += pC[i * 2];
        s[7] += pC[i * 2 + 1];
    }
    __shared__ float red[256];
    __shared__ float tot[8];
    for (int j = 0; j < 8; ++j) {
        red[tid] = s[j];
        __syncthreads();
        for (int st = 128; st > 0; st >>= 1) {
            if (tid < st) red[tid] += red[tid + st];
            __syncthreads();
        }
        if (tid == 0) tot[j] = red[0];
        __syncthreads();
    }
    if (tid == 0) {
        float nTot    = (float)(BB * NN);
        float L_att   = tot[6] / nTot;
        float L_rep   = tot[7] / nTot;
        float nObj    = fmaxf(scalB[1], 1.0f);
        float L_bObj  = scalB[0] / nObj;
        float L_noise = tot[0] / fmaxf(tot[1], 1.0f);
        float pS      = fmaxf(tot[2], 1e-6f);
        float L_e = tot[3] / pS, L_p = tot[4] / pS, L_t = tot[5] / pS;
        out[0] = L_att + L_rep + (L_bObj + L_noise) + L_e + L_p + L_t;
    }
}

extern "C" void kernel_launch(void* const* d_in, const int* in_sizes, int n_in,
                              void* d_out, int out_size, void* d_ws, size_t ws_size,
                              hipStream_t stream) {
    const float* beta = (const float*)d_in[0];
    const float* cc   = (const float*)d_in[1];
    const float* pe   = (const float*)d_in[2];
    const float* pp   = (const float*)d_in[3];
    const float* pt   = (const float*)d_in[4];
    const float* te   = (const float*)d_in[5];
    const float* tp   = (const float*)d_in[6];
    const float* tt   = (const float*)d_in[7];
    const int*   tidx = (const int*)d_in[8];

    char* ws = (char*)d_ws;
    unsigned long long* packed = (unsigned long long*)ws;        // 512 * 8 = 4096 B
    float* objX  = (float*)(ws + 4096);                          // 512 f
    float* objY  = (float*)(ws + 4096 + 2048);                   // 512 f
    float* objQ  = (float*)(ws + 4096 + 4096);                   // 512 f
    float* scalB = (float*)(ws + 4096 + 6144);                   // 2 f
    float* pA    = (float*)(ws + 4096 + 6144 + 64);
    int nA = (BB * NN + 255) / 256;                              // 1563 blocks
    float* pC = pA + (size_t)nA * 6;
    int nC = BB * TB;                                            // 4 * 782 blocks

    k_init <<<2, 256, 0, stream>>>(packed);
    k_hits <<<nA, 256, 0, stream>>>(beta, pe, pp, pt, te, tp, tt, tidx, packed, pA);
    k_alpha<<<1, 512, 0, stream>>>(beta, cc, packed, objX, objY, objQ, scalB);
    k_pair <<<nC, 256, 0, stream>>>(beta, cc, tidx, objX, objY, objQ, pC);
    k_final<<<1, 256, 0, stream>>>(pA, nA, pC, nC, scalB, (float*)d_out);
}